// WavKANClassifier_987842478658
// MI455X (gfx1250) — compile-verified
//
#include <hip/hip_runtime.h>
#include <hip/hip_bf16.h>

// Problem constants (from reference)
#define BATCH 32
#define KIN   6144
#define HDIM  512
#define NCLS  5

typedef __attribute__((ext_vector_type(2))) float v2f;
typedef __attribute__((ext_vector_type(8))) float v8f;

// ---------------------------------------------------------------------------
// Transpose x (B, KIN) -> xT (KIN, B) so the wavelet kernel always reads the
// activation batch-vector as 32 contiguous floats (8x b128 loads per lane).
// ---------------------------------------------------------------------------
__global__ void transpose_x_kernel(const float* __restrict__ x,
                                   float* __restrict__ xT) {
    int idx = blockIdx.x * 256 + threadIdx.x;            // over B*KIN
    if (idx < BATCH * KIN) {
        int b = idx / KIN;
        int i = idx - b * KIN;
        xT[(size_t)i * BATCH + b] = x[idx];
    }
}

// ---------------------------------------------------------------------------
// One wavelet layer: out[o, b] = silu( sum_i w[o,i] * psi(b,o,i) )
//   psi = (1 - s^2) * exp(-s^2/2),  s = (h[i,b] - t[o,i]) / (scale(sr)+1e-8)
// Block = one output row o (512 blocks), 256 threads strided over i.
// Per (o,i): sigmoid/scale computed ONCE, reused across all 32 batches held
// in 32 VGPR accumulators. hin is (IN, 32) so the batch loop is 8x float4.
// ---------------------------------------------------------------------------
template<int IN>
__global__ void wav_layer_kernel(const float* __restrict__ hin,  // (IN, 32)
                                 const float* __restrict__ w,    // (512, IN)
                                 const float* __restrict__ t,    // (512, IN)
                                 const float* __restrict__ sr,   // (512, IN)
                                 float* __restrict__ out)        // (512, 32)
{
    const int o    = blockIdx.x;
    const int tid  = threadIdx.x;      // 0..255
    const int lane = tid & 31;
    const int wv   = tid >> 5;         // 0..7

    float acc[BATCH];
#pragma unroll
    for (int b = 0; b < BATCH; ++b) acc[b] = 0.0f;

    const float* __restrict__ wrow = w  + (size_t)o * IN;
    const float* __restrict__ trow = t  + (size_t)o * IN;
    const float* __restrict__ rrow = sr + (size_t)o * IN;

    for (int i = tid; i < IN; i += 256) {
        // Prefetch next parameter chunk into cache (global_prefetch_b8).
        if (i + 256 < IN) {
            __builtin_prefetch(wrow + i + 256, 0, 0);
            __builtin_prefetch(trow + i + 256, 0, 0);
            __builtin_prefetch(rrow + i + 256, 0, 0);
        }
        const float wval = wrow[i];
        const float tval = trow[i];
        const float rval = rrow[i];
        // scale = 0.01 + 9.99 * sigmoid(sr); inv = 1/(scale + 1e-8)
        const float sig = 1.0f / (1.0f + __expf(-rval));
        const float inv = 1.0f / (0.01f + 9.99f * sig + 1e-8f);

        const float4* __restrict__ hp =
            (const float4*)(hin + (size_t)i * BATCH);
#pragma unroll
        for (int q = 0; q < 8; ++q) {
            float4 hv = hp[q];
            float h4[4] = {hv.x, hv.y, hv.z, hv.w};
#pragma unroll
            for (int j = 0; j < 4; ++j) {
                const float s  = (h4[j] - tval) * inv;
                const float s2 = s * s;
                const float psi = (1.0f - s2) * __expf(-0.5f * s2);
                acc[q * 4 + j] = fmaf(psi, wval, acc[q * 4 + j]);
            }
        }
    }

    // Wave32 reduction: leave the cross-lane sum of acc[b] in lane b.
    float mine = 0.0f;
#pragma unroll
    for (int b = 0; b < BATCH; ++b) {
        float v = acc[b];
#pragma unroll
        for (int off = 16; off > 0; off >>= 1)
            v += __shfl_xor(v, off, 32);
        if (lane == b) mine = v;
    }

    // Cross-wave combine (8 waves) in LDS, then fused SiLU + store.
    __shared__ float red[8][BATCH];
    red[wv][lane] = mine;
    __syncthreads();
    if (tid < BATCH) {
        float v = 0.0f;
#pragma unroll
        for (int k = 0; k < 8; ++k) v += red[k][tid];
        const float y = v / (1.0f + __expf(-v));   // silu
        out[(size_t)o * BATCH + tid] = y;
    }
}

// ---------------------------------------------------------------------------
// LayerNorm over the 512 outputs for each batch row, in place on (512, 32).
// One block per batch element; 256 threads -> 2 elements each.
// ---------------------------------------------------------------------------
__global__ void ln_kernel(float* __restrict__ h,          // (512, 32)
                          const float* __restrict__ g,    // (512)
                          const float* __restrict__ beta) // (512)
{
    const int b    = blockIdx.x;    // 0..31
    const int tid  = threadIdx.x;   // 0..255
    const int lane = tid & 31;
    const int wv   = tid >> 5;

    const float v0 = h[(size_t)tid * BATCH + b];
    const float v1 = h[(size_t)(tid + 256) * BATCH + b];
    float s = v0 + v1;
    float q = v0 * v0 + v1 * v1;
#pragma unroll
    for (int off = 16; off > 0; off >>= 1) {
        s += __shfl_xor(s, off, 32);
        q += __shfl_xor(q, off, 32);
    }
    __shared__ float ss[8], qq[8];
    if (lane == 0) { ss[wv] = s; qq[wv] = q; }
    __syncthreads();
    if (tid == 0) {
        float S = 0.0f, Q = 0.0f;
#pragma unroll
        for (int k = 0; k < 8; ++k) { S += ss[k]; Q += qq[k]; }
        ss[0] = S; qq[0] = Q;
    }
    __syncthreads();
    const float m   = ss[0] * (1.0f / 512.0f);
    const float var = qq[0] * (1.0f / 512.0f) - m * m;
    const float r   = rsqrtf(var + 1e-5f);
    h[(size_t)tid * BATCH + b]         = (v0 - m) * r * g[tid]       + beta[tid];
    h[(size_t)(tid + 256) * BATCH + b] = (v1 - m) * r * g[tid + 256] + beta[tid + 256];
}

// ---------------------------------------------------------------------------
// Classifier head via V_WMMA_F32_16X16X4_F32:
//   out[b, c] = sum_o h[o, b] * cw[c, o] + cb[c]
// M = 32 batches (2 waves x 16-row tile), N = 16 (classes padded 5->16),
// K = 512 in 128 steps of 4.
// A 16x4 f32 layout (ISA 7.12.2): lanes 0-15 M=lane, K={k0,k0+1};
//                                 lanes 16-31 M=lane-16, K={k0+2,k0+3}.
// B 4x16 f32 mirrors A: lane half selects K pair, N = lane&15.
// C/D 8 VGPRs: vgpr r -> M = r + 8*(lane>=16), N = lane&15.
// ---------------------------------------------------------------------------
__global__ void cls_wmma_kernel(const float* __restrict__ h,   // (512, 32)
                                const float* __restrict__ cw,  // (5, 512)
                                const float* __restrict__ cb,  // (5)
                                float* __restrict__ out)       // (32, 5)
{
    const int tid  = threadIdx.x;   // 0..63
    const int wave = tid >> 5;      // M-tile: batches [wave*16, wave*16+16)
    const int lane = tid & 31;
    const int half = lane >> 4;     // selects K pair within the 4-step
    const int l16  = lane & 15;     // A-row (M) and B-col (N)

    const int   ncl  = (l16 < NCLS) ? l16 : (NCLS - 1);
    const float nmsk = (l16 < NCLS) ? 1.0f : 0.0f;
    const int   brow = wave * 16 + l16;

    v8f d = {};
    for (int k0 = 0; k0 < HDIM; k0 += 4) {
        const int ka = k0 + half * 2;
        v2f a, bm;
        a.x  = h[(size_t)ka * BATCH + brow];
        a.y  = h[(size_t)(ka + 1) * BATCH + brow];
        bm.x = cw[(size_t)ncl * HDIM + ka]     * nmsk;
        bm.y = cw[(size_t)ncl * HDIM + ka + 1] * nmsk;
        d = __builtin_amdgcn_wmma_f32_16x16x4_f32(
                /*neg_a=*/false, a, /*neg_b=*/false, bm,
                /*c_mod=*/(short)0, d, /*reuse_a=*/false, /*reuse_b=*/false);
    }

    const float bias = cb[ncl];
#pragma unroll
    for (int r = 0; r < 8; ++r) {
        const int M = r + half * 8;
        const int b = wave * 16 + M;
        if (l16 < NCLS) out[b * NCLS + l16] = d[r] + bias;
    }
}

// ---------------------------------------------------------------------------
// Launch
// ---------------------------------------------------------------------------
extern "C" void kernel_launch(void* const* d_in, const int* in_sizes, int n_in,
                              void* d_out, int out_size, void* d_ws, size_t ws_size,
                              hipStream_t stream) {
    const float* x  = (const float*)d_in[0];
    const float* w0 = (const float*)d_in[1];
    const float* t0 = (const float*)d_in[2];
    const float* s0 = (const float*)d_in[3];
    const float* g0 = (const float*)d_in[4];
    const float* b0 = (const float*)d_in[5];
    const float* w1 = (const float*)d_in[6];
    const float* t1 = (const float*)d_in[7];
    const float* s1 = (const float*)d_in[8];
    const float* g1 = (const float*)d_in[9];
    const float* b1 = (const float*)d_in[10];
    const float* w2 = (const float*)d_in[11];
    const float* t2 = (const float*)d_in[12];
    const float* s2 = (const float*)d_in[13];
    const float* g2 = (const float*)d_in[14];
    const float* b2 = (const float*)d_in[15];
    const float* cw = (const float*)d_in[16];
    const float* cb = (const float*)d_in[17];
    float* out = (float*)d_out;

    float* xT = (float*)d_ws;                 // KIN*32 floats = 786 KB
    float* h0 = xT + (size_t)KIN * BATCH;     // 512*32 floats = 64 KB
    float* h1 = h0 + (size_t)HDIM * BATCH;    // 512*32 floats = 64 KB

    transpose_x_kernel<<<(BATCH * KIN + 255) / 256, 256, 0, stream>>>(x, xT);

    wav_layer_kernel<KIN> <<<HDIM, 256, 0, stream>>>(xT, w0, t0, s0, h0);
    ln_kernel             <<<BATCH, 256, 0, stream>>>(h0, g0, b0);

    wav_layer_kernel<HDIM><<<HDIM, 256, 0, stream>>>(h0, w1, t1, s1, h1);
    ln_kernel             <<<BATCH, 256, 0, stream>>>(h1, g1, b1);

    wav_layer_kernel<HDIM><<<HDIM, 256, 0, stream>>>(h1, w2, t2, s2, h0);
    ln_kernel             <<<BATCH, 256, 0, stream>>>(h0, g2, b2);

    cls_wmma_kernel<<<1, 64, 0, stream>>>(h0, cw, cb, out);
}